// SimpleSSMLayer_45131516346593
// MI455X (gfx1250) — compile-verified
//
#include <hip/hip_runtime.h>
#include <hip/hip_bf16.h>
#include <cmath>

// ---------------------------------------------------------------------------
// SSM layer: y = (scan(a, u@b^T)) @ c^T + u @ d^T   with a = tanh(a_raw)
// fp32 end-to-end; matmuls on the CDNA5 WMMA pipe (v_wmma_f32_16x16x4_f32).
// Weights staged in LDS (conflict-free padded layout); per-wave 16x128 tile;
// whole-chunk A-fragment preload so WMMAs only wait on LDS.
// ---------------------------------------------------------------------------

typedef __attribute__((ext_vector_type(2))) float v2f;
typedef __attribute__((ext_vector_type(4))) float v4f;
typedef __attribute__((ext_vector_type(8))) float v8f;

#define BLOCK_M   128          // 8 waves * 16 rows
#define BLOCK_N   128          // 8 n-tiles of 16
#define NTILES      8
#define KC        256          // compile-time K (Din = S = Dout = 256)
#define KCHUNK     64          // K staged per LDS round
#define KSTEPS    (KCHUNK / 4)
#define LDSW      (KCHUNK + 4) // 68 words: stride%64==4 -> conflict-free b64
#define LDS_ELEMS (BLOCK_N * LDSW)   // 34,816 B

// Cooperative stage of W[n0..n0+127][kc..kc+63] into LDS (row-major, padded).
__device__ __forceinline__
void stage_w(float* lds, const float* __restrict__ W, int n0, int kc, int tid) {
  const int row = tid >> 1;            // 0..127
  const int seg = tid & 1;             // 0..1, 32 floats each
  const float* src = W + (size_t)(n0 + row) * KC + kc + seg * 32;
  float* dst = lds + row * LDSW + seg * 32;   // row*272B: multiple of 16
#pragma unroll
  for (int i = 0; i < 8; ++i)
    ((v4f*)dst)[i] = ((const v4f*)src)[i];
}

// acc += A_tile(m0, kc..kc+63) * W_lds^T ; A fragments preloaded for chunk.
__device__ __forceinline__
void mma_chunk(v8f acc[NTILES], const float* __restrict__ arow_kc,
               const float* lds, int l16, int half) {
  v2f afr[KSTEPS];
#pragma unroll
  for (int i = 0; i < KSTEPS; ++i)
    afr[i] = *(const v2f*)(arow_kc + 4 * i);

  const float* lw = lds + l16 * LDSW + 2 * half;
#pragma unroll
  for (int i = 0; i < KSTEPS; ++i) {
#pragma unroll
    for (int j = 0; j < NTILES; ++j) {
      v2f bf = *(const v2f*)(lw + j * 16 * LDSW + 4 * i);
      acc[j] = __builtin_amdgcn_wmma_f32_16x16x4_f32(
          false, afr[i], false, bf, (short)0, acc[j], false, false);
    }
  }
}

__device__ __forceinline__
void store_tile(float* __restrict__ D, const v8f acc[NTILES],
                int m0, int n0, int N, int l16, int half) {
#pragma unroll
  for (int j = 0; j < NTILES; ++j) {
    const int col = n0 + j * 16 + l16;
#pragma unroll
    for (int r = 0; r < 8; ++r) {
      const int row = m0 + r + 8 * half;   // C layout: lanes16-31 -> M=8+r
      D[(size_t)row * N + col] = acc[j][r];
    }
  }
}

// D(M,N) = A(M,KC) * W(N,KC)^T
__global__ __launch_bounds__(256)
void gemm_nt_wmma(const float* __restrict__ A, const float* __restrict__ W,
                  float* __restrict__ D, int N) {
  __shared__ float lds[LDS_ELEMS];
  const int tid  = threadIdx.x;
  const int lane = tid & 31, wave = tid >> 5;
  const int l16  = lane & 15, half = lane >> 4;
  const int m0 = blockIdx.x * BLOCK_M + wave * 16;
  const int n0 = blockIdx.y * BLOCK_N;
  const float* arow = A + (size_t)(m0 + l16) * KC + 2 * half;

  v8f acc[NTILES];
#pragma unroll
  for (int j = 0; j < NTILES; ++j) acc[j] = 0.0f;

#pragma unroll
  for (int kc = 0; kc < KC; kc += KCHUNK) {
    if (kc) __syncthreads();
    stage_w(lds, W, n0, kc, tid);
    __syncthreads();
    mma_chunk(acc, arow + kc, lds, l16, half);
  }
  store_tile(D, acc, m0, n0, N, l16, half);
}

// D(M,N) = A1*W1^T + A2*W2^T  (fused: single accumulator, single store)
__global__ __launch_bounds__(256)
void gemm_nt_wmma_dual(const float* __restrict__ A1, const float* __restrict__ W1,
                       const float* __restrict__ A2, const float* __restrict__ W2,
                       float* __restrict__ D, int N) {
  __shared__ float lds[LDS_ELEMS];
  const int tid  = threadIdx.x;
  const int lane = tid & 31, wave = tid >> 5;
  const int l16  = lane & 15, half = lane >> 4;
  const int m0 = blockIdx.x * BLOCK_M + wave * 16;
  const int n0 = blockIdx.y * BLOCK_N;
  const float* a1row = A1 + (size_t)(m0 + l16) * KC + 2 * half;
  const float* a2row = A2 + (size_t)(m0 + l16) * KC + 2 * half;

  v8f acc[NTILES];
#pragma unroll
  for (int j = 0; j < NTILES; ++j) acc[j] = 0.0f;

#pragma unroll
  for (int kc = 0; kc < KC; kc += KCHUNK) {
    if (kc) __syncthreads();
    stage_w(lds, W1, n0, kc, tid);
    __syncthreads();
    mma_chunk(acc, a1row + kc, lds, l16, half);
  }
#pragma unroll
  for (int kc = 0; kc < KC; kc += KCHUNK) {
    __syncthreads();
    stage_w(lds, W2, n0, kc, tid);
    __syncthreads();
    mma_chunk(acc, a2row + kc, lds, l16, half);
  }
  store_tile(D, acc, m0, n0, N, l16, half);
}

// ---------------------------------------------------------------------------
// Chunked associative scan for x_t = a*x_{t-1} + bu_t  (diagonal a per s).
// ---------------------------------------------------------------------------
#define S_DIM  256
#define CHUNKS  64

__global__ __launch_bounds__(256)
void ssm_chunk_carry(const float* __restrict__ Bu, const float* __restrict__ a_raw,
                     float* __restrict__ carry, int clen) {
  const int b  = blockIdx.x / CHUNKS;
  const int ch = blockIdx.x % CHUNKS;
  const int s  = threadIdx.x;
  const float a = tanhf(a_raw[s]);
  const float* p = Bu + ((size_t)blockIdx.x * clen) * S_DIM + s;
  float x = 0.0f;
  for (int t = 0; t < clen; ++t)
    x = fmaf(a, x, p[(size_t)t * S_DIM]);
  carry[((size_t)b * CHUNKS + ch) * S_DIM + s] = x;
}

__global__ __launch_bounds__(256)
void ssm_chunk_prefix(const float* __restrict__ a_raw,
                      float* __restrict__ carry, int clen) {
  const int b = blockIdx.x;
  const int s = threadIdx.x;
  const float a = tanhf(a_raw[s]);
  float aL = 1.0f;
  for (int i = 0; i < clen; ++i) aL *= a;
  float P = 0.0f;
  for (int ch = 0; ch < CHUNKS; ++ch) {
    const size_t idx = ((size_t)b * CHUNKS + ch) * S_DIM + s;
    const float cr = carry[idx];
    carry[idx] = P;
    P = fmaf(aL, P, cr);
  }
}

__global__ __launch_bounds__(256)
void ssm_chunk_apply(float* __restrict__ Bu, const float* __restrict__ a_raw,
                     const float* __restrict__ prefix, int clen) {
  const int b  = blockIdx.x / CHUNKS;
  const int ch = blockIdx.x % CHUNKS;
  const int s  = threadIdx.x;
  const float a = tanhf(a_raw[s]);
  float x = prefix[((size_t)b * CHUNKS + ch) * S_DIM + s];
  float* p = Bu + ((size_t)blockIdx.x * clen) * S_DIM + s;
  for (int t = 0; t < clen; ++t) {
    x = fmaf(a, x, p[(size_t)t * S_DIM]);
    p[(size_t)t * S_DIM] = x;
  }
}

// ---------------------------------------------------------------------------
extern "C" void kernel_launch(void* const* d_in, const int* in_sizes, int n_in,
                              void* d_out, int out_size, void* d_ws, size_t ws_size,
                              hipStream_t stream) {
  const float* u     = (const float*)d_in[0];  // (B,T,Din)
  const float* a_raw = (const float*)d_in[1];  // (S,)
  const float* bW    = (const float*)d_in[2];  // (S,Din)
  const float* cW    = (const float*)d_in[3];  // (Dout,S)
  const float* dW    = (const float*)d_in[4];  // (Dout,Din)
  float* y = (float*)d_out;

  const int S    = in_sizes[1];                 // 256
  const int Din  = in_sizes[2] / S;             // 256
  const int Dout = in_sizes[3] / S;             // 256
  const int M    = in_sizes[0] / Din;           // B*T = 65536
  const int Bsz  = 16;
  const int T    = M / Bsz;                     // 4096
  const int clen = T / CHUNKS;                  // 64

  float* Bu    = (float*)d_ws;                  // (M, S) fp32, 64 MB
  float* carry = y;                             // borrow d_out (1 MB); fully
                                                // consumed before final GEMM

  dim3 blk(256);
  dim3 ggemm_bu(M / BLOCK_M, S / BLOCK_N);
  dim3 ggemm_y (M / BLOCK_M, Dout / BLOCK_N);

  // 1) Bu = u @ b^T
  gemm_nt_wmma<<<ggemm_bu, blk, 0, stream>>>(u, bW, Bu, S);

  // 2) chunked scan: Bu -> xs (in place)
  ssm_chunk_carry <<<Bsz * CHUNKS, blk, 0, stream>>>(Bu, a_raw, carry, clen);
  ssm_chunk_prefix<<<Bsz,          blk, 0, stream>>>(a_raw, carry, clen);
  ssm_chunk_apply <<<Bsz * CHUNKS, blk, 0, stream>>>(Bu, a_raw, carry, clen);

  // 3) y = xs @ c^T + u @ d^T  (fused, single store)
  gemm_nt_wmma_dual<<<ggemm_y, blk, 0, stream>>>(Bu, cW, u, dW, y, Dout);
}